// GraphGenerator_68874095558901
// MI455X (gfx1250) — compile-verified
//
#include <hip/hip_runtime.h>
#include <math.h>

#define NEG_SLOPE 0.2f

typedef __attribute__((ext_vector_type(2))) float v2f;
typedef __attribute__((ext_vector_type(8))) float v8f;

__device__ __forceinline__ float leaky(float x) { return x > 0.f ? x : NEG_SLOPE * x; }

__device__ __forceinline__ void atomicMaxF(float* addr, float val) {
    int* ia = (int*)addr;
    int old = __float_as_int(*addr);
    while (__int_as_float(old) < val) {
        int assumed = old;
        old = atomicCAS(ia, assumed, __float_as_int(val));
        if (old == assumed) break;
    }
}

// ---------------- fill / concat ----------------
__global__ void k_fill(float* __restrict__ p, float v, long n) {
    long i = (long)blockIdx.x * blockDim.x + threadIdx.x;
    if (i < n) p[i] = v;
}

__global__ void k_concat(const float* __restrict__ xg, const float* __restrict__ cf,
                         float* __restrict__ h, long ng128, long n128) {
    long i = (long)blockIdx.x * blockDim.x + threadIdx.x;
    if (i >= n128) return;
    h[i] = (i < ng128) ? xg[i] : cf[i - ng128];
}

// ---------------- WMMA fp32 GEMM: C[rows,fout] = A[rows,K] @ W[K,fout] (+bias)(+leaky) ----
// One wave per 16x16 tile. rows must be a multiple of 16 (102048 = 6378*16).
// f32 16x16x4 fragment layout (ISA 7.12.2):
//   A (16x4): lane&15 = M row; lanes 0-15 hold K=0,1 in v0,v1; lanes 16-31 hold K=2,3
//   B (4x16): lane&15 = N col; lanes 0-15 hold K=0,1 rows; lanes 16-31 hold K=2,3
//   C (16x16): VGPR r: lanes 0-15 -> M=r, lanes 16-31 -> M=8+r; N = lane&15
__global__ void k_gemm_wmma(const float* __restrict__ A, const float* __restrict__ W,
                            const float* __restrict__ bias, float* __restrict__ C,
                            int K, int fout, int act) {
    int tilesPerRow = fout >> 4;
    int tile = blockIdx.x;
    int tr = tile / tilesPerRow;
    int tc = tile - tr * tilesPerRow;
    int lane = threadIdx.x;
    int half = lane >> 4;   // 0: K pair {0,1}, 1: K pair {2,3}
    int l15  = lane & 15;

    const float* arow = A + (long)(tr * 16 + l15) * K;
    const float* bcol = W + (tc * 16 + l15);

    v8f acc = {};
    for (int k0 = 0; k0 < K; k0 += 4) {
        int ka = k0 + half * 2;
        v2f a, b;
        a.x = arow[ka];
        a.y = arow[ka + 1];
        b.x = bcol[(long)ka * fout];
        b.y = bcol[(long)(ka + 1) * fout];
        acc = __builtin_amdgcn_wmma_f32_16x16x4_f32(false, a, false, b,
                                                    (short)0, acc, false, false);
    }

    int col = tc * 16 + l15;
    float bv = bias ? bias[col] : 0.f;
    int row0 = tr * 16 + half * 8;
#pragma unroll
    for (int r = 0; r < 8; ++r) {
        float v = acc[r] + bv;
        if (act) v = leaky(v);
        C[(long)(row0 + r) * fout + col] = v;
    }
}

// ---------------- attention coefficients: al_s/al_d [N,4] ----------------
__global__ void k_attn(const float* __restrict__ hW, const float* __restrict__ a_src,
                       const float* __restrict__ a_dst, float* __restrict__ al_s,
                       float* __restrict__ al_d, int N) {
    int i = blockIdx.x * blockDim.x + threadIdx.x;  // n*4 + head
    if (i >= N * 4) return;
    int n = i >> 2, hh = i & 3;
    const float* row = hW + (long)n * 128 + hh * 32;
    const float* as  = a_src + hh * 32;
    const float* ad  = a_dst + hh * 32;
    float s = 0.f, d = 0.f;
#pragma unroll
    for (int j = 0; j < 32; ++j) { float v = row[j]; s += v * as[j]; d += v * ad[j]; }
    al_s[i] = s; al_d[i] = d;
}

// ---------------- edge passes (E real edges + N self loops) ----------------
__device__ __forceinline__ void edge_sd(const int* ei, int E, int e, int& s, int& d) {
    if (e < E) { s = ei[e]; d = ei[E + e]; } else { s = d = e - E; }
}

__global__ void k_edge_max(const int* __restrict__ ei, int E, int ET,
                           const float* __restrict__ al_s, const float* __restrict__ al_d,
                           float* __restrict__ m) {
    int e = blockIdx.x * blockDim.x + threadIdx.x;
    if (e >= ET) return;
    int s, d; edge_sd(ei, E, e, s, d);
#pragma unroll
    for (int hh = 0; hh < 4; ++hh) {
        float v = leaky(al_s[s * 4 + hh] + al_d[d * 4 + hh]);
        atomicMaxF(&m[d * 4 + hh], v);
    }
}

__global__ void k_edge_sum(const int* __restrict__ ei, int E, int ET,
                           const float* __restrict__ al_s, const float* __restrict__ al_d,
                           const float* __restrict__ m, float* __restrict__ denom) {
    int e = blockIdx.x * blockDim.x + threadIdx.x;
    if (e >= ET) return;
    int s, d; edge_sd(ei, E, e, s, d);
#pragma unroll
    for (int hh = 0; hh < 4; ++hh) {
        float v  = leaky(al_s[s * 4 + hh] + al_d[d * 4 + hh]);
        float ex = __expf(v - m[d * 4 + hh]);
        atomicAdd(&denom[d * 4 + hh], ex);
    }
}

__global__ void k_edge_msg(const int* __restrict__ ei, int E, int ET,
                           const float* __restrict__ al_s, const float* __restrict__ al_d,
                           const float* __restrict__ m, const float* __restrict__ denom,
                           const float* __restrict__ hW, float* __restrict__ agg) {
    int idx = blockIdx.x * blockDim.x + threadIdx.x;  // e*4 + head
    if (idx >= ET * 4) return;
    int e = idx >> 2, hh = idx & 3;
    int s, d; edge_sd(ei, E, e, s, d);
    float v     = leaky(al_s[s * 4 + hh] + al_d[d * 4 + hh]);
    float ex    = __expf(v - m[d * 4 + hh]);
    float alpha = ex / (denom[d * 4 + hh] + 1e-16f);
    const float* srcp = hW + (long)s * 128 + hh * 32;
    float*       dstp = agg + (long)d * 128 + hh * 32;
#pragma unroll
    for (int j = 0; j < 32; ++j) atomicAdd(&dstp[j], srcp[j] * alpha);
}

__global__ void k_bias_leaky(float* __restrict__ hb, const float* __restrict__ b, long n128) {
    long i = (long)blockIdx.x * blockDim.x + threadIdx.x;
    if (i >= n128) return;
    hb[i] = leaky(hb[i] + b[i & 127]);
}

// ---------------- MLP output (64 -> 1), ReLU6 ----------------
__global__ void k_mlp_out(const float* __restrict__ hid, const float* __restrict__ w3,
                          const float* __restrict__ b3, float* __restrict__ logits, int N) {
    int n = blockIdx.x * blockDim.x + threadIdx.x;
    if (n >= N) return;
    float acc = b3[0];
    const float* row = hid + (long)n * 64;
#pragma unroll
    for (int j = 0; j < 64; ++j) acc += row[j] * w3[j];
    logits[n] = fminf(fmaxf(acc, 0.f), 6.f);
}

// ---------------- masked argmax (first-occurrence, deterministic) ----------------
// maskMode 0: valid iff i < maskThresh ; maskMode 1: valid iff i != *excludeIdx
__global__ void k_argmax_part(const float* __restrict__ vals, int n, int maskMode,
                              int maskThresh, const int* __restrict__ excludeIdx,
                              float* __restrict__ pV, int* __restrict__ pI) {
    __shared__ float sv[256];
    __shared__ int   si[256];
    int t = threadIdx.x;
    int excl = (maskMode == 1) ? *excludeIdx : -1;
    float best = -__builtin_huge_valf();
    int bi = 0x7fffffff;
    for (int i = blockIdx.x * blockDim.x + t; i < n; i += gridDim.x * blockDim.x) {
        bool ok = (maskMode == 0) ? (i < maskThresh) : (i != excl);
        float v = ok ? vals[i] : -__builtin_huge_valf();
        if (v > best || (v == best && i < bi)) { best = v; bi = i; }
    }
    sv[t] = best; si[t] = bi;
    __syncthreads();
    for (int s = 128; s > 0; s >>= 1) {
        if (t < s && (sv[t + s] > sv[t] || (sv[t + s] == sv[t] && si[t + s] < si[t]))) {
            sv[t] = sv[t + s]; si[t] = si[t + s];
        }
        __syncthreads();
    }
    if (t == 0) { pV[blockIdx.x] = sv[0]; pI[blockIdx.x] = si[0]; }
}

__global__ void k_argmax_final(const float* __restrict__ pV, const int* __restrict__ pI,
                               int B, int* __restrict__ outNode) {
    __shared__ float sv[256];
    __shared__ int   si[256];
    int t = threadIdx.x;
    float best = -__builtin_huge_valf();
    int bi = 0x7fffffff;
    for (int i = t; i < B; i += 256) {
        float v = pV[i]; int idx = pI[i];
        if (v > best || (v == best && idx < bi)) { best = v; bi = idx; }
    }
    sv[t] = best; si[t] = bi;
    __syncthreads();
    for (int s = 128; s > 0; s >>= 1) {
        if (t < s && (sv[t + s] > sv[t] || (sv[t + s] == sv[t] && si[t + s] < si[t]))) {
            sv[t] = sv[t + s]; si[t] = si[t + s];
        }
        __syncthreads();
    }
    if (t == 0) *outNode = si[0];
}

__global__ void k_onehot(float* __restrict__ out, const int* __restrict__ node, int n) {
    int i = blockIdx.x * blockDim.x + threadIdx.x;
    if (i < n) out[i] = (i == *node) ? 1.f : 0.f;
}

static inline int nblk(long n, int b) { return (int)((n + b - 1) / b); }

extern "C" void kernel_launch(void* const* d_in, const int* in_sizes, int n_in,
                              void* d_out, int out_size, void* d_ws, size_t ws_size,
                              hipStream_t stream) {
    const float* xg = (const float*)d_in[0];
    const float* cf = (const float*)d_in[1];
    const int*   ei = (const int*)d_in[2];

    const int NG = in_sizes[0] / 128;
    const int NC = in_sizes[1] / 128;
    const int N  = NG + NC;          // 102048, multiple of 16
    const int E  = in_sizes[2] / 2;  // 1,600,000
    const int ET = E + N;            // edges + self loops

    // workspace layout (floats); agg aliases h (h is dead after the GEMM per layer)
    float* ws   = (float*)d_ws;
    size_t fN128 = (size_t)N * 128;
    float* h    = ws;                 // [N,128] node features / aggregation buffer
    float* hW   = h + fN128;          // [N,128] h@W  (also MLP hidden1 [N,64] + hidden2 [N,64])
    float* als  = hW + fN128;         // [N,4]
    float* ald  = als + (size_t)N * 4;
    float* mb   = ald + (size_t)N * 4; // segment max [N,4]
    float* db   = mb  + (size_t)N * 4; // segment sum [N,4]
    float* pV   = db  + (size_t)N * 4; // 512 argmax partial values
    int*   pI   = (int*)(pV + 512);    // 512 argmax partial indices
    int*   node = pI + 512;            // node[0]=start, node[1]=end

    const int T = 256;
    const long n128 = (long)N * 128;
    const long n4   = (long)N * 4;
    const dim3 wave(32);

    k_concat<<<nblk(n128, T), T, 0, stream>>>(xg, cf, h, (long)NG * 128, n128);

    // ---- 3 GAT layers ----
    for (int L = 0; L < 3; ++L) {
        const float* W  = (const float*)d_in[3 + L * 4 + 0];
        const float* as = (const float*)d_in[3 + L * 4 + 1];
        const float* ad = (const float*)d_in[3 + L * 4 + 2];
        const float* bb = (const float*)d_in[3 + L * 4 + 3];

        k_gemm_wmma<<<(N / 16) * (128 / 16), wave, 0, stream>>>(h, W, nullptr, hW, 128, 128, 0);
        k_attn<<<nblk(n4, T), T, 0, stream>>>(hW, as, ad, als, ald, N);

        k_fill<<<nblk(n128, T), T, 0, stream>>>(h, 0.f, n128);                       // agg = 0
        k_fill<<<nblk(n4, T), T, 0, stream>>>(mb, -__builtin_huge_valf(), n4);
        k_fill<<<nblk(n4, T), T, 0, stream>>>(db, 0.f, n4);

        k_edge_max<<<nblk(ET, T), T, 0, stream>>>(ei, E, ET, als, ald, mb);
        k_edge_sum<<<nblk(ET, T), T, 0, stream>>>(ei, E, ET, als, ald, mb, db);
        k_edge_msg<<<nblk((long)ET * 4, T), T, 0, stream>>>(ei, E, ET, als, ald, mb, db, hW, h);
        k_bias_leaky<<<nblk(n128, T), T, 0, stream>>>(h, bb, n128);
    }

    float* out = (float*)d_out;
    float* hid1 = hW;                       // [N,64]
    float* hid2 = hW + (size_t)N * 64;      // [N,64]

    // ---- start MLP + argmax + one-hot ----
    {
        const float* w1 = (const float*)d_in[15]; const float* b1 = (const float*)d_in[16];
        const float* w2 = (const float*)d_in[17]; const float* b2 = (const float*)d_in[18];
        const float* w3 = (const float*)d_in[19]; const float* b3 = (const float*)d_in[20];
        k_gemm_wmma<<<(N / 16) * (64 / 16), wave, 0, stream>>>(h, w1, b1, hid1, 128, 64, 1);
        k_gemm_wmma<<<(N / 16) * (64 / 16), wave, 0, stream>>>(hid1, w2, b2, hid2, 64, 64, 1);
        k_mlp_out<<<nblk(N, T), T, 0, stream>>>(hid2, w3, b3, out, N);
    }
    k_argmax_part<<<512, 256, 0, stream>>>(out, N, 0, NG, nullptr, pV, pI);
    k_argmax_final<<<1, 256, 0, stream>>>(pV, pI, 512, &node[0]);
    k_onehot<<<nblk(N, T), T, 0, stream>>>(out + N, &node[0], N);

    // ---- end MLP + argmax + one-hot ----
    {
        const float* w1 = (const float*)d_in[21]; const float* b1 = (const float*)d_in[22];
        const float* w2 = (const float*)d_in[23]; const float* b2 = (const float*)d_in[24];
        const float* w3 = (const float*)d_in[25]; const float* b3 = (const float*)d_in[26];
        k_gemm_wmma<<<(N / 16) * (64 / 16), wave, 0, stream>>>(h, w1, b1, hid1, 128, 64, 1);
        k_gemm_wmma<<<(N / 16) * (64 / 16), wave, 0, stream>>>(hid1, w2, b2, hid2, 64, 64, 1);
        k_mlp_out<<<nblk(N, T), T, 0, stream>>>(hid2, w3, b3, out + 2L * N, N);
    }
    k_argmax_part<<<512, 256, 0, stream>>>(out + 2L * N, N, 1, 0, &node[0], pV, pI);
    k_argmax_final<<<1, 256, 0, stream>>>(pV, pI, 512, &node[1]);
    k_onehot<<<nblk(N, T), T, 0, stream>>>(out + 3L * N, &node[1], N);

    (void)n_in; (void)out_size; (void)ws_size;
}